// GravNetModel_mod_74758200754562
// MI455X (gfx1250) — compile-verified
//
#include <hip/hip_runtime.h>
#include <math.h>

// GravNet forward for MI455X (gfx1250, wave32).
// B=4 events x V=2048 hits, K=64 neighbours, f32 end-to-end.
// Dense layers use V_WMMA_F32_16X16X4_F32; W panel staged in LDS (transposed,
// padded); K-loop is a 2-stage software pipeline with two statically named
// register sets (no rotation copies) so loads for one set overlap the WMMA
// burst of the other.

#define BEV 4
#define VEV 2048
#define NROW (BEV * VEV)
#define KNN_K 64

typedef __attribute__((ext_vector_type(2))) float v2f;
typedef __attribute__((ext_vector_type(8))) float v8f;

__device__ __forceinline__ float act_apply(float v, int act) {
  if (act == 1) return v > 0.f ? v : 0.f;          // relu
  if (act == 2) return v > 0.f ? v : expm1f(v);    // elu
  return v;
}

__device__ __forceinline__ v8f wmma_f32(v2f a, v2f b, v8f c) {
  // (neg_a, A, neg_b, B, c_mod, C, reuse_a, reuse_b)
  return __builtin_amdgcn_wmma_f32_16x16x4_f32(false, a, false, b,
                                               (short)0, c, false, false);
}

// Y[r,c] = act( sum_k X[r,k] * W[k,c] + bias[c] )
// Block = 4 waves; block covers 256 rows x 16 cols. Each wave: 4 row-tiles
// (64 rows) sharing one B stream from LDS. din % 4 == 0 (din >= 32);
// NROW % 256 == 0. Dynamic LDS: 16 * (din+4) floats.
__global__ __launch_bounds__(128)
void wmma_dense_kernel(const float* __restrict__ X, int ldx,
                       const float* __restrict__ W,
                       const float* __restrict__ bias,
                       float* __restrict__ Y, int ldy,
                       int din, int dout, int act) {
  extern __shared__ float lwb[];               // [16][din+4], transposed W panel
  const int dinp = din + 4;                    // pad keeps LDS banks spread
  const int tid  = threadIdx.y * 32 + threadIdx.x;
  const int n0   = blockIdx.x * 16;

  // Stage W[:, n0:n0+16] transposed; zero-fill out-of-range columns so the
  // compute loop needs no predication (EXEC stays all-ones for WMMA).
  {
    const int c   = tid & 15;
    const int col = n0 + c;
    const bool cv = col < dout;
    for (int k = tid >> 4; k < din; k += 8)
      lwb[c * dinp + k] = cv ? W[(size_t)k * dout + col] : 0.f;
  }
  __syncthreads();

  const int lane = threadIdx.x;
  const int l    = lane & 15;
  const int half = lane >> 4;                  // 0 or 1

  const int rowBase = blockIdx.y * 256 + threadIdx.y * 64;
  // A: lane l holds row (tileBase + l); half 0 -> K={0,1}, half 1 -> K={2,3}.
  const float* xr0 = X + (size_t)(rowBase + l) * ldx + 2 * half;
  const float* xr1 = xr0 + (size_t)16 * ldx;
  const float* xr2 = xr1 + (size_t)16 * ldx;
  const float* xr3 = xr2 + (size_t)16 * ldx;
  // B: lane l holds column n0+l; contiguous (k, k+1) pair -> ds_load_b64.
  const float* wb = lwb + l * dinp + 2 * half;

  v8f acc0 = {}, acc1 = {}, acc2 = {}, acc3 = {};

  // Two statically named pipeline register sets (no rotation copies).
  v2f s0b, s0a0, s0a1, s0a2, s0a3;
  v2f s1b, s1a0, s1a1, s1a2, s1a3;

#define LOAD_SET(S, K)                                                      \
  do {                                                                      \
    S##b.x  = wb[(K)];      S##b.y  = wb[(K) + 1];                          \
    S##a0.x = xr0[(K)];     S##a0.y = xr0[(K) + 1];                         \
    S##a1.x = xr1[(K)];     S##a1.y = xr1[(K) + 1];                         \
    S##a2.x = xr2[(K)];     S##a2.y = xr2[(K) + 1];                         \
    S##a3.x = xr3[(K)];     S##a3.y = xr3[(K) + 1];                         \
  } while (0)

#define MAC_SET(S)                                                          \
  do {                                                                      \
    acc0 = wmma_f32(S##a0, S##b, acc0);                                     \
    acc1 = wmma_f32(S##a1, S##b, acc1);                                     \
    acc2 = wmma_f32(S##a2, S##b, acc2);                                     \
    acc3 = wmma_f32(S##a3, S##b, acc3);                                     \
  } while (0)

  LOAD_SET(s0, 0);
  int k = 0;
  for (; k + 8 < din; k += 8) {   // both prefetches below stay in-range
    LOAD_SET(s1, k + 4);
    MAC_SET(s0);
    LOAD_SET(s0, k + 8);
    MAC_SET(s1);
  }
  if (k + 4 < din) {              // two steps remain: k and k+4
    LOAD_SET(s1, k + 4);
    MAC_SET(s0);
    MAC_SET(s1);
  } else {                        // one step remains: k
    MAC_SET(s0);
  }
#undef LOAD_SET
#undef MAC_SET

  const int col = n0 + l;
  if (col < dout) {
    const float bv = bias[col];
    const int r0 = rowBase + 8 * half;         // D VGPR r -> row tile + r + 8*half
    #pragma unroll
    for (int r = 0; r < 8; ++r) {
      Y[(size_t)(r0 +  0 + r) * ldy + col] = act_apply(acc0[r] + bv, act);
      Y[(size_t)(r0 + 16 + r) * ldy + col] = act_apply(acc1[r] + bv, act);
      Y[(size_t)(r0 + 32 + r) * ldy + col] = act_apply(acc2[r] + bv, act);
      Y[(size_t)(r0 + 48 + r) * ldy + col] = act_apply(acc3[r] + bv, act);
    }
  }
}

// Self-inclusive KNN within each event in 8-d coord space.
// One 256-thread block per query hit: d2 row in LDS, then 64 argmin rounds.
__global__ __launch_bounds__(256)
void knn_kernel(const float* __restrict__ coords,
                int* __restrict__ knn_idx, float* __restrict__ knn_d2) {
  __shared__ float d2s[VEV];
  __shared__ float rmin[256];
  __shared__ int   ridx[256];

  const int v = blockIdx.x;
  const int e = v / VEV;
  const int tid = threadIdx.x;

  float q[8];
  #pragma unroll
  for (int t = 0; t < 8; ++t) q[t] = coords[(size_t)v * 8 + t];

  const float* cb = coords + (size_t)e * VEV * 8;
  for (int w = tid; w < VEV; w += 256) {
    const float* cw = cb + (size_t)w * 8;
    float d = 0.f;
    #pragma unroll
    for (int t = 0; t < 8; ++t) { float df = q[t] - cw[t]; d += df * df; }
    d2s[w] = d;
  }
  __syncthreads();

  for (int kk = 0; kk < KNN_K; ++kk) {
    float best = INFINITY; int bi = VEV;
    for (int w = tid; w < VEV; w += 256) {
      float d = d2s[w];
      if (d < best || (d == best && w < bi)) { best = d; bi = w; }
    }
    rmin[tid] = best; ridx[tid] = bi;
    __syncthreads();
    for (int s = 128; s > 0; s >>= 1) {
      if (tid < s) {
        if (rmin[tid + s] < rmin[tid] ||
            (rmin[tid + s] == rmin[tid] && ridx[tid + s] < ridx[tid])) {
          rmin[tid] = rmin[tid + s]; ridx[tid] = ridx[tid + s];
        }
      }
      __syncthreads();
    }
    if (tid == 0) {
      knn_idx[(size_t)v * KNN_K + kk] = ridx[0];
      knn_d2 [(size_t)v * KNN_K + kk] = rmin[0];
      d2s[ridx[0]] = INFINITY;
    }
    __syncthreads();
  }
}

// f[v] = [ feats[v] | mean_k(w_k*feats[nbr_k]) | max_k(w_k*feats[nbr_k]) ],
// w_k = exp(-10*d2_k). One block per hit, one thread per channel (128).
__global__ __launch_bounds__(128)
void aggregate_kernel(const float* __restrict__ feats,
                      const int* __restrict__ knn_idx,
                      const float* __restrict__ knn_d2,
                      float* __restrict__ f) {
  const int v = blockIdx.x;
  const int c = threadIdx.x;
  const int base = (v / VEV) * VEV;
  const int*   idx = knn_idx + (size_t)v * KNN_K;
  const float* dd  = knn_d2  + (size_t)v * KNN_K;

  float s = 0.f, mx = -INFINITY;
  for (int k = 0; k < KNN_K; ++k) {
    const int j = base + idx[k];           // wave-uniform load
    const float wgt = expf(-10.f * dd[k]); // wave-uniform load
    const float val = feats[(size_t)j * 128 + c] * wgt;
    s += val;
    mx = fmaxf(mx, val);
  }
  f[(size_t)v * 384 + c]        = feats[(size_t)v * 128 + c];
  f[(size_t)v * 384 + 128 + c]  = s * (1.f / KNN_K);
  f[(size_t)v * 384 + 256 + c]  = mx;
}

// Per-event channel means: mean[e*C + c] = mean over V rows of x.
__global__ __launch_bounds__(128)
void event_mean_kernel(const float* __restrict__ x, int ldx, int C,
                       float* __restrict__ mean) {
  const int e = blockIdx.x;
  const int c = threadIdx.x;
  if (c >= C) return;
  const float* p = x + (size_t)e * VEV * ldx + c;
  float s = 0.f;
  for (int v = 0; v < VEV; ++v) s += p[(size_t)v * ldx];
  mean[e * C + c] = s * (1.f / VEV);
}

// dst[v] = [ x[v] | mean[event(v)] ]
__global__ __launch_bounds__(256)
void ge_concat_kernel(const float* __restrict__ x, int ldx, int C,
                      const float* __restrict__ mean,
                      float* __restrict__ dst, int ldd) {
  const int i = blockIdx.x * blockDim.x + threadIdx.x;
  if (i >= NROW * C) return;
  const int v = i / C, c = i % C;
  const int e = v / VEV;
  dst[(size_t)v * ldd + c]     = x[(size_t)v * ldx + c];
  dst[(size_t)v * ldd + C + c] = mean[e * C + c];
}

// dst[v, 0:C] = src[v, 0:C] with independent strides.
__global__ __launch_bounds__(256)
void copy_cols_kernel(const float* __restrict__ src, int lds_, int C,
                      float* __restrict__ dst, int ldd) {
  const int i = blockIdx.x * blockDim.x + threadIdx.x;
  if (i >= NROW * C) return;
  const int v = i / C, c = i % C;
  dst[(size_t)v * ldd + c] = src[(size_t)v * lds_ + c];
}

struct Lin { const float* b; const float* w; };

extern "C" void kernel_launch(void* const* d_in, const int* in_sizes, int n_in,
                              void* d_out, int out_size, void* d_ws, size_t ws_size,
                              hipStream_t stream) {
  (void)in_sizes; (void)n_in; (void)out_size; (void)ws_size;

  const float* feat = (const float*)d_in[0];
  // d_in[1] = row_splits (uniform, baked in), d_in[2] = test_arr (unused).

  // Params flattened in jax pytree (sorted-key) order:
  // blocks[0..3]{dn1,dn2,dn3,feat,out,spatial}{b,w}, dense_start{b,w}, odn1..odn5{b,w}
  int pi = 3;
  Lin blk[4][6];
  for (int i = 0; i < 4; ++i)
    for (int j = 0; j < 6; ++j) {
      blk[i][j].b = (const float*)d_in[pi++];
      blk[i][j].w = (const float*)d_in[pi++];
    }
  Lin dense_start; dense_start.b = (const float*)d_in[pi++]; dense_start.w = (const float*)d_in[pi++];
  Lin odn[5];
  for (int i = 0; i < 5; ++i) { odn[i].b = (const float*)d_in[pi++]; odn[i].w = (const float*)d_in[pi++]; }
  enum { J_DN1 = 0, J_DN2 = 1, J_DN3 = 2, J_FEAT = 3, J_OUT = 4, J_SPATIAL = 5 };

  // Workspace carve-up (256B aligned).
  char* wsp = (char*)d_ws;
  auto alloc = [&](size_t elems) -> float* {
    float* p = (float*)wsp;
    wsp += ((elems * sizeof(float) + 255) / 256) * 256;
    return p;
  };
  float* XCAT   = alloc((size_t)NROW * 464);  // final concat buffer; slices written in place
  float* ge     = alloc((size_t)NROW * 192);
  float* meanb  = alloc((size_t)BEV * 256);
  float* coords = alloc((size_t)NROW * 8);
  float* featsb = alloc((size_t)NROW * 128);
  float* fcat   = alloc((size_t)NROW * 384);
  float* o196   = alloc((size_t)NROW * 196);
  float* t128   = alloc((size_t)NROW * 128);
  float* t96    = alloc((size_t)NROW * 96);
  float* h1     = alloc((size_t)NROW * 128);
  float* h2     = alloc((size_t)NROW * 128);
  int*   knn_i  = (int*)alloc((size_t)NROW * KNN_K);
  float* knn_d  = alloc((size_t)NROW * KNN_K);

  auto gemm = [&](const float* X, int ldx, Lin L, float* Y, int ldy,
                  int din, int dout, int act) {
    dim3 block(32, 4);                            // 4 waves; 256 rows x 16 cols per block
    dim3 grid((dout + 15) / 16, NROW / 256);
    size_t shbytes = (size_t)16 * (din + 4) * sizeof(float);
    wmma_dense_kernel<<<grid, block, shbytes, stream>>>(X, ldx, L.w, L.b, Y, ldy, din, dout, act);
  };
  auto ge_pair = [&](const float* x, int ldx, int C, float* dst, int ldd) {
    event_mean_kernel<<<dim3(BEV), dim3(128), 0, stream>>>(x, ldx, C, meanb);
    int tot = NROW * C;
    ge_concat_kernel<<<dim3((tot + 255) / 256), dim3(256), 0, stream>>>(x, ldx, C, meanb, dst, ldd);
  };

  // --- stem ---
  // XCAT[:, 0:16] = feat
  {
    int tot = NROW * 16;
    copy_cols_kernel<<<dim3((tot + 255) / 256), dim3(256), 0, stream>>>(feat, 16, 16, XCAT, 464);
  }
  ge_pair(feat, 16, 16, ge, 32);                         // [N,32]
  gemm(ge, 32, dense_start, XCAT + 16, 464, 32, 64, 2);  // elu -> XCAT[:,16:80]

  // --- 4 GravNet blocks ---
  int xoff = 16, din = 64;
  for (int i = 0; i < 4; ++i) {
    const float* xin = XCAT + xoff;
    gemm(xin, 464, blk[i][J_SPATIAL], coords, 8, din, 8, 0);     // coords [N,8]
    gemm(xin, 464, blk[i][J_FEAT],    featsb, 128, din, 128, 1); // feats relu [N,128]
    knn_kernel<<<dim3(NROW), dim3(256), 0, stream>>>(coords, knn_i, knn_d);
    aggregate_kernel<<<dim3(NROW), dim3(128), 0, stream>>>(featsb, knn_i, knn_d, fcat);
    gemm(fcat, 384, blk[i][J_OUT], o196, 196, 384, 196, 1);      // [N,196] relu
    gemm(o196, 196, blk[i][J_DN1], t128, 128, 196, 128, 1);      // dn1 relu
    gemm(t128, 128, blk[i][J_DN2], t96,  96, 128, 96, 1);        // dn2 relu
    ge_pair(t96, 96, 96, ge, 192);                               // [N,192]
    const int oc = 80 + 96 * i;
    gemm(ge, 192, blk[i][J_DN3], XCAT + oc, 464, 192, 96, 1);    // dn3 relu -> XCAT slice
    xoff = oc; din = 96;
  }

  // --- output head on the 464-wide concat ---
  gemm(XCAT, 464, odn[0], h1, 128, 464, 128, 1);  // relu
  gemm(h1, 128, odn[1], h2, 128, 128, 128, 2);    // elu
  gemm(h2, 128, odn[2], h1, 128, 128, 128, 2);    // elu
  gemm(h1, 128, odn[3], h2, 128, 128, 128, 1);    // relu
  gemm(h2, 128, odn[4], (float*)d_out, 128, 128, 128, 1); // relu -> [N,128]
}